// MyModel_61933428415408
// MI455X (gfx1250) — compile-verified
//
#include <hip/hip_runtime.h>
#include <hip/hip_bf16.h>

// rsqrt(x) over N fp32 elements. Memory-bound streamer:
//  - 128-bit NT loads/stores (global_load_b128 / global_store_b128, th:NT)
//  - v_rsq_f32 + 1 Newton-Raphson refinement
//  - 32-bit loop induction (N fits), addresses as base + (size_t)idx so the
//    backend can use saddr + 32-bit voffset instead of 64-bit VALU adds
//  - grid-stride, unrolled x4 for 4 in-flight b128 loads per thread

typedef __attribute__((ext_vector_type(4))) float f32x4;

__device__ __forceinline__ float rsqrt_refined(float x) {
    // v_rsq_f32 seed, then one NR step: y' = y * (1.5 - 0.5*x*y*y)
    float y = __builtin_amdgcn_rsqf(x);
    float t = __builtin_fmaf(-0.5f * x * y, y, 1.5f);
    return y * t;
}

__device__ __forceinline__ f32x4 rsqrt4(f32x4 v) {
    f32x4 r;
    r.x = rsqrt_refined(v.x);
    r.y = rsqrt_refined(v.y);
    r.z = rsqrt_refined(v.z);
    r.w = rsqrt_refined(v.w);
    return r;
}

__device__ __forceinline__ f32x4 ld_nt(const f32x4* p, unsigned idx) {
    return __builtin_nontemporal_load(p + (size_t)idx);
}
__device__ __forceinline__ void st_nt(f32x4 v, f32x4* p, unsigned idx) {
    __builtin_nontemporal_store(v, p + (size_t)idx);
}

__global__ void __launch_bounds__(256)
rsqrt_stream_kernel(const f32x4* __restrict__ x, f32x4* __restrict__ out,
                    unsigned nvec,
                    const float* __restrict__ xs, float* __restrict__ outs,
                    unsigned tail_base, unsigned tail_count) {
    const unsigned stride = gridDim.x * blockDim.x;
    unsigned i = blockIdx.x * blockDim.x + threadIdx.x;

    // Main body: 4 independent 128-bit NT loads in flight per thread.
    unsigned i3;
    while ((i3 = i + 3u * stride) < nvec) {
        f32x4 v0 = ld_nt(x, i);
        f32x4 v1 = ld_nt(x, i + stride);
        f32x4 v2 = ld_nt(x, i + 2u * stride);
        f32x4 v3 = ld_nt(x, i3);
        st_nt(rsqrt4(v0), out, i);
        st_nt(rsqrt4(v1), out, i + stride);
        st_nt(rsqrt4(v2), out, i + 2u * stride);
        st_nt(rsqrt4(v3), out, i3);
        i += 4u * stride;
    }
    // Remainder of the vectorized range.
    for (; i < nvec; i += stride) {
        st_nt(rsqrt4(ld_nt(x, i)), out, i);
    }
    // Scalar tail (N not divisible by 4).
    unsigned tid = blockIdx.x * blockDim.x + threadIdx.x;
    if (tid < tail_count) {
        size_t j = (size_t)tail_base + tid;
        outs[j] = rsqrt_refined(xs[j]);
    }
}

extern "C" void kernel_launch(void* const* d_in, const int* in_sizes, int n_in,
                              void* d_out, int out_size, void* d_ws, size_t ws_size,
                              hipStream_t stream) {
    (void)n_in; (void)d_ws; (void)ws_size; (void)out_size;
    const float* x = (const float*)d_in[0];
    float* out = (float*)d_out;

    const unsigned n = (unsigned)in_sizes[0];
    const unsigned nvec = n >> 2;            // # of float4 chunks (fits 32-bit)
    const unsigned tail_base = nvec << 2;
    const unsigned tail_count = n - tail_base;

    const int block = 256;                   // 8 wave32 waves per workgroup
    const int per_thread_vec = 4;            // unroll factor in the kernel
    unsigned long long want =
        ((unsigned long long)nvec + (unsigned long long)block * per_thread_vec - 1) /
        ((unsigned long long)block * per_thread_vec);
    if (want < 1) want = 1;
    if (want > 1048576ull) want = 1048576ull;
    const int grid = (int)want;              // 16384 blocks for N = 64M

    rsqrt_stream_kernel<<<grid, block, 0, stream>>>(
        (const f32x4*)x, (f32x4*)out, nvec, x, out, tail_base, tail_count);
}